// GlobalPointer_71270687309945
// MI455X (gfx1250) — compile-verified
//
#include <hip/hip_runtime.h>
#include <cstddef>

typedef __attribute__((ext_vector_type(2))) float v2f;
typedef __attribute__((ext_vector_type(8))) float v8f;

#define Bq 8
#define Sq 1024
#define Hq 768
#define Tq 8
#define DIq 64
#define HIDq 128
#define Kq 8
#define HEADSq 2
#define NF 9            /* K+1 nodes */
#define F1 256          /* HEADS*HID */
#define DWN 1024        /* T*2*DI */
#define AUGD 80         /* DI + K*2 gaussian columns */
#define NEGV 1.0e12f

// ---------------------------------------------------------------------------
// Stage 1: node features (global mean + span means)
// ---------------------------------------------------------------------------
__global__ void nodes_kernel(const float* __restrict__ hidden,
                             const int* __restrict__ spans,
                             float* __restrict__ nodes) {
  int b = blockIdx.x / NF;
  int node = blockIdx.x % NF;
  int tid = threadIdx.x;
  int r0, r1;
  if (node == 0) { r0 = 0; r1 = Sq - 1; }
  else {
    const int* sp = spans + (b * Kq + (node - 1)) * 3;
    r0 = sp[0]; r1 = sp[1];
  }
  float inv = 1.0f / (float)(r1 - r0 + 1);
  const float* hb = hidden + (size_t)b * Sq * Hq;
  for (int d = tid; d < Hq; d += blockDim.x) {
    float acc = 0.f;
    for (int r = r0; r <= r1; ++r) acc += hb[(size_t)r * Hq + d];
    nodes[((size_t)b * NF + node) * Hq + d] = acc * inv;
  }
}

// ---------------------------------------------------------------------------
// Stage 2: both GAT layers per batch (tiny: 9 nodes), output enhanced (B,256)
// ---------------------------------------------------------------------------
__global__ void __launch_bounds__(256) gat_kernel(
    const float* __restrict__ nodes, const int* __restrict__ spans,
    const float* __restrict__ W1, const float* __restrict__ a_src1,
    const float* __restrict__ a_dst1, const float* __restrict__ b1,
    const float* __restrict__ ln_g, const float* __restrict__ ln_b,
    const float* __restrict__ W2, const float* __restrict__ a_src2,
    const float* __restrict__ a_dst2, const float* __restrict__ b2,
    float* __restrict__ enhanced) {
  __shared__ float xs[NF * Hq];          // 27648 B
  __shared__ float hs[NF * F1];          // 9216 B
  __shared__ float g1s[NF * F1];         // 9216 B
  __shared__ float adjs[NF * NF];
  __shared__ float asr[NF * HEADSq];
  __shared__ float ads[NF * HEADSq];
  __shared__ float sc[NF * NF * HEADSq];
  int b = blockIdx.x, tid = threadIdx.x;

  // adjacency: eye + hub links + 2*containment (adj[1+a][1+p]=2 if span p contains span a)
  if (tid < NF * NF) {
    int i = tid / NF, j = tid % NF;
    float a = (i == j) ? 1.0f : 0.0f;
    if (i > 0 && j == 0) a += 1.0f;
    if (i == 0 && j > 0) a += 1.0f;
    if (i > 0 && j > 0) {
      int sa = spans[(b * Kq + i - 1) * 3], ea = spans[(b * Kq + i - 1) * 3 + 1];
      int sb = spans[(b * Kq + j - 1) * 3], eb = spans[(b * Kq + j - 1) * 3 + 1];
      bool same = (sa == sb) && (ea == eb);
      if (sb <= sa && ea <= eb && !same) a += 2.0f;
    }
    adjs[tid] = a;
  }
  for (int i = tid; i < NF * Hq; i += 256) xs[i] = nodes[(size_t)b * NF * Hq + i];
  __syncthreads();

  // h = x @ W1  (thread = output column)
  for (int n = 0; n < NF; ++n) {
    float acc = 0.f;
    const float* xr = xs + n * Hq;
    for (int d = 0; d < Hq; ++d) acc += xr[d] * W1[(size_t)d * F1 + tid];
    hs[n * F1 + tid] = acc;
  }
  __syncthreads();
  if (tid < NF * HEADSq) {
    int n = tid >> 1, hd = tid & 1;
    float s1 = 0.f, s2 = 0.f;
    for (int d = 0; d < HIDq; ++d) {
      float hv = hs[n * F1 + hd * HIDq + d];
      s1 += hv * a_src1[hd * HIDq + d];
      s2 += hv * a_dst1[hd * HIDq + d];
    }
    asr[tid] = s1; ads[tid] = s2;
  }
  __syncthreads();
  if (tid < NF * NF * HEADSq) {
    int hd = tid & 1; int ij = tid >> 1; int i = ij / NF, j = ij % NF;
    float e = ads[i * HEADSq + hd] + asr[j * HEADSq + hd];
    e = (e > 0.f) ? e : 0.2f * e;
    float a = adjs[i * NF + j];
    sc[tid] = (a > 0.f) ? (e + __logf(a)) : -1.0e30f;
  }
  __syncthreads();
  if (tid < NF * HEADSq) {   // softmax over source j
    int i = tid >> 1, hd = tid & 1;
    float mx = -1e38f;
    for (int j = 0; j < NF; ++j) mx = fmaxf(mx, sc[((i * NF + j) << 1) + hd]);
    float sm = 0.f;
    for (int j = 0; j < NF; ++j) { float e = __expf(sc[((i * NF + j) << 1) + hd] - mx); sc[((i * NF + j) << 1) + hd] = e; sm += e; }
    float inv = 1.0f / sm;
    for (int j = 0; j < NF; ++j) sc[((i * NF + j) << 1) + hd] *= inv;
  }
  __syncthreads();
  {
    int hd = tid >> 7;
    for (int i = 0; i < NF; ++i) {
      float o = 0.f;
      for (int j = 0; j < NF; ++j) o += sc[((i * NF + j) << 1) + hd] * hs[j * F1 + tid];
      o += b1[tid];
      g1s[i * F1 + tid] = fmaxf(o, 0.0f);
    }
  }
  __syncthreads();
  if (tid < NF) {            // layernorm per node
    float mu = 0.f;
    for (int c = 0; c < F1; ++c) mu += g1s[tid * F1 + c];
    mu /= (float)F1;
    float var = 0.f;
    for (int c = 0; c < F1; ++c) { float dd = g1s[tid * F1 + c] - mu; var += dd * dd; }
    var /= (float)F1;
    float inv = rsqrtf(var + 1e-5f);
    for (int c = 0; c < F1; ++c)
      g1s[tid * F1 + c] = (g1s[tid * F1 + c] - mu) * inv * ln_g[c] + ln_b[c];
  }
  __syncthreads();
  // GAT layer 2 (h2 reuses hs)
  for (int n = 0; n < NF; ++n) {
    float acc = 0.f;
    for (int d = 0; d < F1; ++d) acc += g1s[n * F1 + d] * W2[(size_t)d * F1 + tid];
    hs[n * F1 + tid] = acc;
  }
  __syncthreads();
  if (tid < NF * HEADSq) {
    int n = tid >> 1, hd = tid & 1;
    float s1 = 0.f, s2 = 0.f;
    for (int d = 0; d < HIDq; ++d) {
      float hv = hs[n * F1 + hd * HIDq + d];
      s1 += hv * a_src2[hd * HIDq + d];
      s2 += hv * a_dst2[hd * HIDq + d];
    }
    asr[tid] = s1; ads[tid] = s2;
  }
  __syncthreads();
  if (tid < NF * NF * HEADSq) {
    int hd = tid & 1; int ij = tid >> 1; int i = ij / NF, j = ij % NF;
    float e = ads[i * HEADSq + hd] + asr[j * HEADSq + hd];
    e = (e > 0.f) ? e : 0.2f * e;
    float a = adjs[i * NF + j];
    sc[tid] = (a > 0.f) ? (e + __logf(a)) : -1.0e30f;
  }
  __syncthreads();
  if (tid < NF * HEADSq) {
    int i = tid >> 1, hd = tid & 1;
    float mx = -1e38f;
    for (int j = 0; j < NF; ++j) mx = fmaxf(mx, sc[((i * NF + j) << 1) + hd]);
    float sm = 0.f;
    for (int j = 0; j < NF; ++j) { float e = __expf(sc[((i * NF + j) << 1) + hd] - mx); sc[((i * NF + j) << 1) + hd] = e; sm += e; }
    float inv = 1.0f / sm;
    for (int j = 0; j < NF; ++j) sc[((i * NF + j) << 1) + hd] *= inv;
  }
  __syncthreads();
  {
    int hd = tid >> 7;
    float mean = 0.f;
    for (int i = 0; i < NF; ++i) {
      float o = 0.f;
      for (int j = 0; j < NF; ++j) o += sc[((i * NF + j) << 1) + hd] * hs[j * F1 + tid];
      o += b2[tid];
      mean += fmaxf(o, 0.0f);
    }
    enhanced[b * F1 + tid] = mean / (float)NF;
  }
}

// ---------------------------------------------------------------------------
// Stage 3: eW[b,n] = enhanced[b,:] @ dense_W[768:,:]   (folds rank-1 term)
// ---------------------------------------------------------------------------
__global__ void ew_kernel(const float* __restrict__ enhanced,
                          const float* __restrict__ dW,
                          float* __restrict__ eW) {
  int b = blockIdx.x;
  int n = blockIdx.y * blockDim.x + threadIdx.x;
  float acc = 0.f;
  for (int j = 0; j < F1; ++j)
    acc += enhanced[b * F1 + j] * dW[(size_t)(Hq + j) * DWN + n];
  eW[b * DWN + n] = acc;
}

// ---------------------------------------------------------------------------
// Stage 4: qk = hidden(8192x768) @ W_top(768x1024) + dense_b + cover*eW
// 64x64 tiles, 8 wave32s, f32 WMMA 16x16x4.
// ---------------------------------------------------------------------------
__global__ void __launch_bounds__(256) dense_wmma_kernel(
    const float* __restrict__ hidden, const float* __restrict__ dW,
    const float* __restrict__ dense_b, const float* __restrict__ eW,
    const int* __restrict__ spans, float* __restrict__ qk) {
  __shared__ float As[64 * 68];
  __shared__ float Bs[64 * 68];
  __shared__ int sSt[Kq], sEn[Kq];
  int m0 = blockIdx.x * 64;
  int n0 = blockIdx.y * 64;
  int b = m0 / Sq;               // tiles never cross batch (1024 % 64 == 0)
  int tid = threadIdx.x;
  if (tid < Kq) {
    sSt[tid] = spans[(b * Kq + tid) * 3];
    sEn[tid] = spans[(b * Kq + tid) * 3 + 1];
  }
  int wave = tid >> 5, lane = tid & 31;
  int lm = lane & 15, lk2 = (lane >> 4) << 1;
  int mi0 = wave >> 2, ni0 = wave & 3;   // subtile 0
  int mi1 = mi0 + 2, ni1 = ni0;          // subtile 1
  v8f acc0 = {0.f, 0.f, 0.f, 0.f, 0.f, 0.f, 0.f, 0.f};
  v8f acc1 = {0.f, 0.f, 0.f, 0.f, 0.f, 0.f, 0.f, 0.f};

  for (int k0 = 0; k0 < Hq; k0 += 64) {
    __syncthreads();
    for (int i = tid; i < 1024; i += 256) {
      int r = i >> 4, c4 = i & 15;
      *(float4*)(As + r * 68 + c4 * 4) =
          *(const float4*)(hidden + (size_t)(m0 + r) * Hq + k0 + c4 * 4);
    }
    for (int i = tid; i < 1024; i += 256) {
      int r = i >> 4, c4 = i & 15;
      *(float4*)(Bs + r * 68 + c4 * 4) =
          *(const float4*)(dW + (size_t)(k0 + r) * DWN + n0 + c4 * 4);
    }
    __syncthreads();
    for (int kk = 0; kk < 64; kk += 4) {
      v2f a0 = *(const v2f*)(As + (mi0 * 16 + lm) * 68 + kk + lk2);
      v2f a1 = *(const v2f*)(As + (mi1 * 16 + lm) * 68 + kk + lk2);
      v2f b0, b1v;
      b0.x = Bs[(kk + lk2) * 68 + ni0 * 16 + lm];
      b0.y = Bs[(kk + lk2 + 1) * 68 + ni0 * 16 + lm];
      b1v = b0;  // ni1 == ni0
      acc0 = __builtin_amdgcn_wmma_f32_16x16x4_f32(false, a0, false, b0,
                                                   (short)0, acc0, false, false);
      acc1 = __builtin_amdgcn_wmma_f32_16x16x4_f32(false, a1, false, b1v,
                                                   (short)0, acc1, false, false);
    }
  }
  __syncthreads();
  int lhi = (lane >> 4) << 3;
  for (int v = 0; v < 8; ++v) {
    {
      int gm = m0 + mi0 * 16 + v + lhi, gn = n0 + ni0 * 16 + lm;
      int s = gm & (Sq - 1);
      float cover = 0.f;
      for (int kk = 0; kk < Kq; ++kk) cover += (s >= sSt[kk] && s <= sEn[kk]) ? 1.f : 0.f;
      qk[(size_t)gm * DWN + gn] = acc0[v] + dense_b[gn] + cover * eW[b * DWN + gn];
    }
    {
      int gm = m0 + mi1 * 16 + v + lhi, gn = n0 + ni1 * 16 + lm;
      int s = gm & (Sq - 1);
      float cover = 0.f;
      for (int kk = 0; kk < Kq; ++kk) cover += (s >= sSt[kk] && s <= sEn[kk]) ? 1.f : 0.f;
      qk[(size_t)gm * DWN + gn] = acc1[v] + dense_b[gn] + cover * eW[b * DWN + gn];
    }
  }
}

// ---------------------------------------------------------------------------
// Stage 5a: RoPE -> Qaug/Kaug columns [0,64)
// ---------------------------------------------------------------------------
__global__ void rope_kernel(const float* __restrict__ qk,
                            float* __restrict__ Qaug, float* __restrict__ Kaug) {
  int idx = blockIdx.x * blockDim.x + threadIdx.x;  // B*T*S*32
  int i = idx & 31;
  int s = (idx >> 5) & (Sq - 1);
  int t = (idx >> 15) & 7;
  int b = idx >> 18;
  float inv = __expf(-(2.0f * (float)i / (float)DIq) * 9.2103403719761836f); // 10000^(-2i/DI)
  float ang = (float)s * inv;
  float sn, cs;
  __sincosf(ang, &sn, &cs);
  const float* src = qk + (size_t)(b * Sq + s) * DWN + t * (2 * DIq);
  float q0 = src[2 * i], q1 = src[2 * i + 1];
  float k0 = src[DIq + 2 * i], k1 = src[DIq + 2 * i + 1];
  float* qd = Qaug + ((size_t)(b * Tq + t) * Sq + s) * AUGD;
  float* kd = Kaug + ((size_t)(b * Tq + t) * Sq + s) * AUGD;
  qd[2 * i] = q0 * cs - q1 * sn;
  qd[2 * i + 1] = q1 * cs + q0 * sn;
  kd[2 * i] = k0 * cs - k1 * sn;
  kd[2 * i + 1] = k1 * cs + k0 * sn;
}

// ---------------------------------------------------------------------------
// Stage 5b: gaussian rank-1 columns [64,80): Q side carries corr weight
// ---------------------------------------------------------------------------
__global__ void gauss_kernel(const int* __restrict__ spans,
                             const float* __restrict__ centers,
                             const float* __restrict__ sigmas,
                             const float* __restrict__ wts,
                             const float* __restrict__ corr,
                             float* __restrict__ Qaug, float* __restrict__ Kaug) {
  int idx = blockIdx.x * blockDim.x + threadIdx.x;  // B*T*S
  int s = idx & (Sq - 1);
  int t = (idx >> 10) & 7;
  int b = idx >> 13;
  float* qd = Qaug + ((size_t)(b * Tq + t) * Sq + s) * AUGD + DIq;
  float* kd = Kaug + ((size_t)(b * Tq + t) * Sq + s) * AUGD + DIq;
  for (int k = 0; k < Kq; ++k) {
    int st = spans[(b * Kq + k) * 3];
    int en = spans[(b * Kq + k) * 3 + 1];
    int et = spans[(b * Kq + k) * 3 + 2];
    float w = corr[et * Tq + t];
    for (int c = 0; c < 2; ++c) {
      float sg = sigmas[c];
      float dxs = ((float)s - (float)st - centers[c]) / sg;
      float dxe = ((float)s - (float)en - centers[c]) / sg;
      qd[2 * k + c] = w * wts[c] * __expf(-0.5f * dxs * dxs);
      kd[2 * k + c] = wts[c] * __expf(-0.5f * dxe * dxe);
    }
  }
}

// ---------------------------------------------------------------------------
// Stage 6: logits[bt] = Qaug(1024x80) @ Kaug(1024x80)^T, fused mask/tril/scale
// ---------------------------------------------------------------------------
__global__ void __launch_bounds__(256) logits_wmma_kernel(
    const float* __restrict__ Qaug, const float* __restrict__ Kaug,
    const float* __restrict__ att, float* __restrict__ out) {
  __shared__ float Qs[64 * 84];
  __shared__ float Ks[64 * 84];
  int bt = blockIdx.x;
  int m0 = blockIdx.y * 64;
  int n0 = blockIdx.z * 64;
  int b = bt >> 3;
  int tid = threadIdx.x;
  int wave = tid >> 5, lane = tid & 31;
  int lm = lane & 15, lk2 = (lane >> 4) << 1;
  const float* Qg = Qaug + ((size_t)bt * Sq + m0) * AUGD;
  const float* Kg = Kaug + ((size_t)bt * Sq + n0) * AUGD;
  for (int i = tid; i < 1280; i += 256) {
    int r = i / 20, c4 = i % 20;
    *(float4*)(Qs + r * 84 + c4 * 4) = *(const float4*)(Qg + (size_t)r * AUGD + c4 * 4);
  }
  for (int i = tid; i < 1280; i += 256) {
    int r = i / 20, c4 = i % 20;
    *(float4*)(Ks + r * 84 + c4 * 4) = *(const float4*)(Kg + (size_t)r * AUGD + c4 * 4);
  }
  __syncthreads();
  int mi0 = wave >> 2, ni0 = wave & 3;
  int mi1 = mi0 + 2;
  v8f acc0 = {0.f, 0.f, 0.f, 0.f, 0.f, 0.f, 0.f, 0.f};
  v8f acc1 = {0.f, 0.f, 0.f, 0.f, 0.f, 0.f, 0.f, 0.f};
  for (int kk = 0; kk < AUGD; kk += 4) {
    v2f a0 = *(const v2f*)(Qs + (mi0 * 16 + lm) * 84 + kk + lk2);
    v2f a1 = *(const v2f*)(Qs + (mi1 * 16 + lm) * 84 + kk + lk2);
    v2f bv = *(const v2f*)(Ks + (ni0 * 16 + lm) * 84 + kk + lk2); // B^T: read k-row of Kaug
    acc0 = __builtin_amdgcn_wmma_f32_16x16x4_f32(false, a0, false, bv,
                                                 (short)0, acc0, false, false);
    acc1 = __builtin_amdgcn_wmma_f32_16x16x4_f32(false, a1, false, bv,
                                                 (short)0, acc1, false, false);
  }
  int lhi = (lane >> 4) << 3;
  for (int v = 0; v < 8; ++v) {
    {
      int gm = m0 + mi0 * 16 + v + lhi, gn = n0 + ni0 * 16 + lm;
      float pad = att[b * Sq + gn];
      float val = acc0[v];
      val = val * pad - (1.0f - pad) * NEGV;
      if (gn < gm) val -= NEGV;
      out[((size_t)bt * Sq + gm) * Sq + gn] = val * 0.125f;
    }
    {
      int gm = m0 + mi1 * 16 + v + lhi, gn = n0 + ni0 * 16 + lm;
      float pad = att[b * Sq + gn];
      float val = acc1[v];
      val = val * pad - (1.0f - pad) * NEGV;
      if (gn < gm) val -= NEGV;
      out[((size_t)bt * Sq + gm) * Sq + gn] = val * 0.125f;
    }
  }
}

// ---------------------------------------------------------------------------
extern "C" void kernel_launch(void* const* d_in, const int* in_sizes, int n_in,
                              void* d_out, int out_size, void* d_ws, size_t ws_size,
                              hipStream_t stream) {
  const float* hidden = (const float*)d_in[0];
  const float* att    = (const float*)d_in[1];
  const int*   spans  = (const int*)d_in[2];
  const float* W1     = (const float*)d_in[3];
  const float* a_src1 = (const float*)d_in[4];
  const float* a_dst1 = (const float*)d_in[5];
  const float* b1     = (const float*)d_in[6];
  const float* ln_g   = (const float*)d_in[7];
  const float* ln_b   = (const float*)d_in[8];
  const float* W2     = (const float*)d_in[9];
  const float* a_src2 = (const float*)d_in[10];
  const float* a_dst2 = (const float*)d_in[11];
  const float* b2     = (const float*)d_in[12];
  const float* dense_W = (const float*)d_in[13];
  const float* dense_b = (const float*)d_in[14];
  const float* centers = (const float*)d_in[15];
  const float* sigmas  = (const float*)d_in[16];
  const float* wts     = (const float*)d_in[17];
  const float* corr    = (const float*)d_in[18];
  float* out = (float*)d_out;

  // workspace layout (floats), ~76 MB total
  float* ws = (float*)d_ws;
  float* nodes    = ws;                        // 8*9*768      = 55296
  float* enhanced = ws + 55296;                // 8*256        = 2048
  float* eW       = ws + 57344;                // 8*1024       = 8192
  float* qk       = ws + 65536;                // 8*1024*1024  = 8388608
  float* Qaug     = qk + 8388608;              // 8*8*1024*80  = 5242880
  float* Kaug     = Qaug + 5242880;            // 5242880

  nodes_kernel<<<Bq * NF, 256, 0, stream>>>(hidden, spans, nodes);
  gat_kernel<<<Bq, 256, 0, stream>>>(nodes, spans, W1, a_src1, a_dst1, b1,
                                     ln_g, ln_b, W2, a_src2, a_dst2, b2, enhanced);
  ew_kernel<<<dim3(Bq, DWN / 256), 256, 0, stream>>>(enhanced, dense_W, eW);
  dense_wmma_kernel<<<dim3((Bq * Sq) / 64, DWN / 64), 256, 0, stream>>>(
      hidden, dense_W, dense_b, eW, spans, qk);
  rope_kernel<<<(Bq * Tq * Sq * 32) / 256, 256, 0, stream>>>(qk, Qaug, Kaug);
  gauss_kernel<<<(Bq * Tq * Sq) / 256, 256, 0, stream>>>(spans, centers, sigmas,
                                                         wts, corr, Qaug, Kaug);
  logits_wmma_kernel<<<dim3(Bq * Tq, Sq / 64, Sq / 64), 256, 0, stream>>>(
      Qaug, Kaug, att, out);
}